// ConvTopKSAE_30030411334099
// MI455X (gfx1250) — compile-verified
//
#include <hip/hip_runtime.h>
#include <stdint.h>

typedef float v2f __attribute__((ext_vector_type(2)));
typedef float v8f __attribute__((ext_vector_type(8)));

#define B_      8
#define C_      64
#define HC_     2048
#define NPIX    4096                 // H*W
#define PER_BATCH (HC_*NPIX)         // 8388608
#define K_TOP   167772               // max(1, int(0.02 * 8388608))

// ---- workspace layout (in 32-bit words from d_ws base) ----
#define WS_BETA    0                 // 8 floats
#define WS_PREFIX  8                 // 8 uint
#define WS_KREM    16                // 8 uint
#define WS_THR     24                // 8 floats
#define WS_INVN    32                // 64 floats
#define WS_HIST    96                // 8*256 uint
#define WS_PSUM    2144              // 8192 floats (per-encoder-block partial sums)
#define WS_PSUMSQ  10336             // 8192 floats
#define WS_WDN     18528             // 2048*64 floats, stored transposed [C][HC]

// ---------------- init: zero select state, compute decoder inv-norms ----------------
__global__ void k_init(const float* __restrict__ W_enc, float* __restrict__ ws) {
  unsigned* wsu = (unsigned*)ws;
  int t = threadIdx.x;
  if (t < 8) {
    wsu[WS_PREFIX + t] = 0u;
    wsu[WS_KREM + t]   = (unsigned)K_TOP;
  }
  for (int i = t; i < 8 * 256; i += blockDim.x) wsu[WS_HIST + i] = 0u;
  if (t < C_) {
    float s = 0.f;
    for (int o = 0; o < HC_; ++o) { float w = W_enc[o * C_ + t]; s += w * w; }
    float nrm = sqrtf(s);
    if (nrm < 1e-12f) nrm = 1e-12f;
    ws[WS_INVN + t] = 1.0f / nrm;
  }
}

// ---------------- build normalized transposed decoder weight wdnT[c][o] ----------------
__global__ void k_wdn(const float* __restrict__ W_enc, float* __restrict__ ws) {
  int i = blockIdx.x * blockDim.x + threadIdx.x;   // 131072 elements
  if (i < HC_ * C_) {
    int c = i >> 11;          // / HC_
    int o = i & (HC_ - 1);
    ws[WS_WDN + i] = W_enc[o * C_ + c] * ws[WS_INVN + c];
  }
}

// ---------------- encoder GEMM: pre = W_enc*(x-b_dec)+b_enc, via f32 WMMA ----------------
// grid = 8192 blocks x 256 threads (8 waves). Each wave: 64M x 16N tile, K=64 in steps of 4.
__global__ void __launch_bounds__(256)
k_encoder(const float* __restrict__ x, const float* __restrict__ W_enc,
          const float* __restrict__ b_enc, const float* __restrict__ b_dec,
          float* __restrict__ pre, float* __restrict__ ws) {
  const int lane = threadIdx.x & 31;
  const int wid  = threadIdx.x >> 5;
  const int gw   = blockIdx.x * 8 + wid;     // 65536 waves
  const int b    = gw >> 13;                 // / 8192
  const int t    = gw & 8191;
  const int m0   = (t >> 8) * 64;            // 32 M-macro-tiles
  const int n0   = (t & 255) * 16;           // 256 N-tiles
  const int lrow = lane & 15;
  const int kh   = lane >> 4;

  v8f acc[4];
#pragma unroll
  for (int i = 0; i < 4; ++i)
#pragma unroll
    for (int v = 0; v < 8; ++v) acc[i][v] = 0.0f;

  const float2* w2 = (const float2*)W_enc;

#pragma unroll 4
  for (int kk = 0; kk < C_; kk += 4) {
    const int cA = kk + kh * 2;
    v2f bf;
    bf[0] = x[(b * C_ + cA + 0) * NPIX + n0 + lrow] - b_dec[cA + 0];
    bf[1] = x[(b * C_ + cA + 1) * NPIX + n0 + lrow] - b_dec[cA + 1];
#pragma unroll
    for (int i = 0; i < 4; ++i) {
      const int m = m0 + i * 16 + lrow;
      float2 w = w2[m * 32 + (kk >> 1) + kh];
      v2f af; af[0] = w.x; af[1] = w.y;
      acc[i] = __builtin_amdgcn_wmma_f32_16x16x4_f32(
          false, af, false, bf, (short)0, acc[i], false, false);
    }
  }

  // epilogue: add b_enc, store pre, accumulate per-wave sum / sumsq
  float s = 0.f, s2 = 0.f;
#pragma unroll
  for (int i = 0; i < 4; ++i) {
#pragma unroll
    for (int v = 0; v < 8; ++v) {
      const int row = m0 + i * 16 + v + kh * 8;
      float val = acc[i][v] + b_enc[row];
      pre[(b * HC_ + row) * NPIX + n0 + lrow] = val;
      s += val; s2 += val * val;
    }
  }
#pragma unroll
  for (int off = 16; off > 0; off >>= 1) {
    s  += __shfl_down(s,  off, 32);
    s2 += __shfl_down(s2, off, 32);
  }
  __shared__ float ls[8], ls2[8];
  if (lane == 0) { ls[wid] = s; ls2[wid] = s2; }
  __syncthreads();
  if (threadIdx.x == 0) {
    float bs = 0.f, bs2 = 0.f;
#pragma unroll
    for (int i = 0; i < 8; ++i) { bs += ls[i]; bs2 += ls2[i]; }
    ws[WS_PSUM  + blockIdx.x] = bs;     // deterministic: one slot per block
    ws[WS_PSUMSQ+ blockIdx.x] = bs2;
  }
}

// ---------------- beta: deterministic reduce of 1024 partials per batch ----------------
__global__ void k_beta(float* __restrict__ ws) {
  int b = threadIdx.x;
  if (b >= B_) return;
  double ds = 0.0, ds2 = 0.0;
  for (int i = 0; i < 1024; ++i) {
    ds  += (double)ws[WS_PSUM   + b * 1024 + i];
    ds2 += (double)ws[WS_PSUMSQ + b * 1024 + i];
  }
  const double n = (double)PER_BATCH;
  double var = (ds2 - ds * ds / n) / (n - 1.0);
  if (var < 0.0) var = 0.0;
  ws[WS_BETA + b] = (float)(sqrt(var) * (10.0 + 1e-6));   // 1/eps + 1e-6
}

// ---------------- acts = relu(pre + beta*gumbel) in place + histogram of byte3 ----------------
#define K3_ITEMS 16
__global__ void __launch_bounds__(256)
k_acts_hist(float* __restrict__ acts, const float* __restrict__ gumbel,
            float* __restrict__ ws) {
  __shared__ unsigned hist[256];
  for (int i = threadIdx.x; i < 256; i += blockDim.x) hist[i] = 0u;
  __syncthreads();
  const int blocksPerBatch = PER_BATCH / (256 * K3_ITEMS);   // 2048
  const int b = blockIdx.x / blocksPerBatch;
  const int chunk = blockIdx.x % blocksPerBatch;
  const float beta = ws[WS_BETA + b];
  const long base = (long)b * PER_BATCH + (long)chunk * (256 * K3_ITEMS);
#pragma unroll 4
  for (int i = 0; i < K3_ITEMS; ++i) {
    const long idx = base + threadIdx.x + i * 256;
    float a = acts[idx] + beta * gumbel[idx];
    a = (a > 0.0f) ? a : 0.0f;            // force +0 bits (keeps uint ordering valid)
    acts[idx] = a;
    atomicAdd(&hist[__float_as_uint(a) >> 24], 1u);
  }
  __syncthreads();
  unsigned* gh = &((unsigned*)ws)[WS_HIST + b * 256];
  for (int i = threadIdx.x; i < 256; i += blockDim.x)
    if (hist[i]) atomicAdd(&gh[i], hist[i]);
}

// ---------------- radix-select step: pick bin containing kth-largest, clear hist ----------------
__global__ void k_select(float* __restrict__ ws, int shift) {
  int b = threadIdx.x;
  if (b >= B_) return;
  unsigned* hist = &((unsigned*)ws)[WS_HIST + b * 256];
  unsigned* pfx  = &((unsigned*)ws)[WS_PREFIX];
  unsigned* krem = &((unsigned*)ws)[WS_KREM];
  unsigned kr = krem[b];
  unsigned acc = 0, chosen = 0, newkr = kr;
  for (int bin = 255; bin >= 0; --bin) {
    unsigned c = hist[bin];
    if (acc + c >= kr) { chosen = (unsigned)bin; newkr = kr - acc; break; }
    acc += c;
  }
  unsigned p = pfx[b] | (chosen << shift);
  pfx[b] = p;
  krem[b] = newkr;
  for (int i = 0; i < 256; ++i) hist[i] = 0u;
  if (shift == 0) ws[WS_THR + b] = __uint_as_float(p);
}

// ---------------- histogram pass over candidates matching current prefix ----------------
#define KP_ITEMS 32
__global__ void __launch_bounds__(256)
k_hist_pass(const float* __restrict__ acts, float* __restrict__ ws, int shift) {
  __shared__ unsigned hist[256];
  for (int i = threadIdx.x; i < 256; i += blockDim.x) hist[i] = 0u;
  __syncthreads();
  const int blocksPerBatch = PER_BATCH / (256 * KP_ITEMS);   // 1024
  const int b = blockIdx.x / blocksPerBatch;
  const int chunk = blockIdx.x % blocksPerBatch;
  const unsigned hshift = (unsigned)(shift + 8);
  const unsigned want = ((unsigned*)ws)[WS_PREFIX + b] >> hshift;
  const long base = (long)b * PER_BATCH + (long)chunk * (256 * KP_ITEMS);
#pragma unroll 4
  for (int i = 0; i < KP_ITEMS; ++i) {
    unsigned bits = __float_as_uint(acts[base + threadIdx.x + i * 256]);
    if ((bits >> hshift) == want)
      atomicAdd(&hist[(bits >> shift) & 255u], 1u);
  }
  __syncthreads();
  unsigned* gh = &((unsigned*)ws)[WS_HIST + b * 256];
  for (int i = threadIdx.x; i < 256; i += blockDim.x)
    if (hist[i]) atomicAdd(&gh[i], hist[i]);
}

// ---------------- apply top-k mask in place ----------------
#define KM_ITEMS 16
__global__ void __launch_bounds__(256)
k_mask(float* __restrict__ acts, const float* __restrict__ ws) {
  const int blocksPerBatch = PER_BATCH / (256 * KM_ITEMS);   // 2048
  const int b = blockIdx.x / blocksPerBatch;
  const int chunk = blockIdx.x % blocksPerBatch;
  const float thr = ws[WS_THR + b];
  const long base = (long)b * PER_BATCH + (long)chunk * (256 * KM_ITEMS);
#pragma unroll 4
  for (int i = 0; i < KM_ITEMS; ++i) {
    const long idx = base + threadIdx.x + i * 256;
    float a = acts[idx];
    acts[idx] = (a >= thr) ? a : 0.0f;
  }
}

// ---------------- decoder GEMM: recon = wdnT * sparse + b_dec, via f32 WMMA ----------------
// grid = 256 blocks x 256 threads (8 waves). Each wave: full 64M x 16N strip, K=2048 in steps of 4.
__global__ void __launch_bounds__(256)
k_decoder(const float* __restrict__ sparse, const float* __restrict__ b_dec,
          const float* __restrict__ ws, float* __restrict__ recon) {
  const int lane = threadIdx.x & 31;
  const int wid  = threadIdx.x >> 5;
  const int gw   = blockIdx.x * 8 + wid;     // 2048 waves
  const int b    = gw >> 8;                  // 256 N-tiles per batch
  const int n0   = (gw & 255) * 16;
  const int lrow = lane & 15;
  const int kh   = lane >> 4;

  v8f acc[4];
#pragma unroll
  for (int i = 0; i < 4; ++i)
#pragma unroll
    for (int v = 0; v < 8; ++v) acc[i][v] = 0.0f;

  const float2* w2 = (const float2*)(ws + WS_WDN);   // wdnT[c][o], row length HC_

#pragma unroll 2
  for (int kk = 0; kk < HC_; kk += 4) {
    const int o = kk + kh * 2;
    const long sb = ((long)b * HC_ + o) * NPIX + n0 + lrow;
    v2f bf;
    bf[0] = sparse[sb];
    bf[1] = sparse[sb + NPIX];
#pragma unroll
    for (int i = 0; i < 4; ++i) {
      const int c = i * 16 + lrow;
      float2 w = w2[c * (HC_ / 2) + (kk >> 1) + kh];
      v2f af; af[0] = w.x; af[1] = w.y;
      acc[i] = __builtin_amdgcn_wmma_f32_16x16x4_f32(
          false, af, false, bf, (short)0, acc[i], false, false);
    }
  }

#pragma unroll
  for (int i = 0; i < 4; ++i) {
#pragma unroll
    for (int v = 0; v < 8; ++v) {
      const int c = i * 16 + v + kh * 8;
      recon[((long)b * C_ + c) * NPIX + n0 + lrow] = acc[i][v] + b_dec[c];
    }
  }
}

// ---------------- launch ----------------
extern "C" void kernel_launch(void* const* d_in, const int* in_sizes, int n_in,
                              void* d_out, int out_size, void* d_ws, size_t ws_size,
                              hipStream_t stream) {
  const float* x      = (const float*)d_in[0];
  const float* W_enc  = (const float*)d_in[1];
  const float* b_enc  = (const float*)d_in[2];
  const float* b_dec  = (const float*)d_in[3];
  const float* gumbel = (const float*)d_in[4];

  float* recon  = (float*)d_out;                       // [8,64,64,64]
  float* sparse = recon + (size_t)B_ * C_ * NPIX;      // [8,2048,64,64]; holds pre->acts->sparse
  float* ws     = (float*)d_ws;

  k_init   <<<1,    256, 0, stream>>>(W_enc, ws);
  k_wdn    <<<512,  256, 0, stream>>>(W_enc, ws);
  k_encoder<<<8192, 256, 0, stream>>>(x, W_enc, b_enc, b_dec, sparse, ws);
  k_beta   <<<1,    8,   0, stream>>>(ws);
  k_acts_hist<<<16384, 256, 0, stream>>>(sparse, gumbel, ws);
  k_select <<<1, 8, 0, stream>>>(ws, 24);
  k_hist_pass<<<8192, 256, 0, stream>>>(sparse, ws, 16);
  k_select <<<1, 8, 0, stream>>>(ws, 16);
  k_hist_pass<<<8192, 256, 0, stream>>>(sparse, ws, 8);
  k_select <<<1, 8, 0, stream>>>(ws, 8);
  k_hist_pass<<<8192, 256, 0, stream>>>(sparse, ws, 0);
  k_select <<<1, 8, 0, stream>>>(ws, 0);
  k_mask   <<<16384, 256, 0, stream>>>(sparse, ws);
  k_decoder<<<256,  256, 0, stream>>>(sparse, b_dec, ws, recon);
}